// GAT_70557722738796
// MI455X (gfx1250) — compile-verified
//
#include <hip/hip_runtime.h>
#include <math.h>

// ---------------------------------------------------------------------------
// GAT (2-layer, PyG-style) for MI455X / gfx1250.
// Dense GEMMs use V_WMMA_F32_16X16X4_F32 (full fp32, matches reference
// precision; GEMM cost is negligible vs the ~2 GB of edge traffic, which at
// 23.3 TB/s HBM is the real roofline). Edge softmax/aggregation uses native
// fp32 global atomics; segment-max uses a monotonic uint encoding so it maps
// to a single GLOBAL_ATOMIC_MAX_U32.
// ---------------------------------------------------------------------------

typedef __attribute__((ext_vector_type(2))) float v2f;
typedef __attribute__((ext_vector_type(8))) float v8f;

constexpr int Nn   = 50000;          // nodes  (multiple of 16 -> clean tiles)
constexpr int Ee   = 1600000;        // raw edges
constexpr int Eall = Ee + Nn;        // + self loops
constexpr int H1 = 8, F1 = 8, HID = 64, OUT = 64, IN = 256;

// -------------------------------- helpers ----------------------------------

__device__ __forceinline__ void edge_ends(const int* __restrict__ ei, int e,
                                          int& s, int& d) {
  if (e < Ee) { s = ei[e]; d = ei[Ee + e]; }
  else        { s = e - Ee; d = s; }          // self loop
}

__device__ __forceinline__ float leaky(float a) {
  return a >= 0.0f ? a : 0.2f * a;
}

// monotonic float<->uint map: f0 < f1  <=>  enc(f0) < enc(f1)
__device__ __forceinline__ unsigned fenc(float f) {
  unsigned u = __float_as_uint(f);
  return (u & 0x80000000u) ? ~u : (u | 0x80000000u);
}
__device__ __forceinline__ float fdec(unsigned u) {
  return __uint_as_float((u & 0x80000000u) ? (u & 0x7fffffffu) : ~u);
}

__global__ __launch_bounds__(256) void zero_f32(float* __restrict__ p, int n) {
  int i = blockIdx.x * 256 + threadIdx.x;
  if (i < n) p[i] = 0.0f;
}

// ------------------------- fp32 WMMA GEMM (M x K)*(K x 64) -----------------
// One wave per 16-row tile; 4 v8f accumulators cover all 64 output columns.
// B is staged through LDS in 64-row chunks (16 KB).
// VGPR layouts per CDNA5 ISA 7.12.2:
//   A 16x4 : lane%16 = M,  K = vgpr + 2*(lane/16)
//   B 4x16 : lane%16 = N,  K = vgpr + 2*(lane/16)
//   C 16x16: lane%16 = N,  M = vgpr + 8*(lane/16)
template <int K>
__global__ __launch_bounds__(256) void gemm_wmma_f32(
    const float* __restrict__ A, const float* __restrict__ B,
    float* __restrict__ C, int nrows) {
  __shared__ float Bs[64 * 64];
  const int lane  = threadIdx.x & 31;
  const int wave  = threadIdx.x >> 5;
  const int mrow  = lane & 15;
  const int khalf = lane >> 4;
  const int m0    = (blockIdx.x * 8 + wave) * 16;   // wave-uniform
  v8f acc0 = {}, acc1 = {}, acc2 = {}, acc3 = {};

  for (int kc = 0; kc < K; kc += 64) {
    __syncthreads();
    for (int i = threadIdx.x; i < 64 * 64; i += 256)
      Bs[i] = B[(size_t)(kc + (i >> 6)) * 64 + (i & 63)];
    __syncthreads();
    if (m0 < nrows) {                 // wave-uniform guard: EXEC stays all-1s
      for (int k = 0; k < 64; k += 4) {
        v2f a = *(const v2f*)&A[(size_t)(m0 + mrow) * K + kc + k + 2 * khalf];
        const float* brow0 = &Bs[(k + 2 * khalf) * 64 + mrow];
        const float* brow1 = brow0 + 64;
        v2f b0, b1, b2, b3;
        b0.x = brow0[0];  b0.y = brow1[0];
        b1.x = brow0[16]; b1.y = brow1[16];
        b2.x = brow0[32]; b2.y = brow1[32];
        b3.x = brow0[48]; b3.y = brow1[48];
        acc0 = __builtin_amdgcn_wmma_f32_16x16x4_f32(false, a, false, b0,
                                                     (short)0, acc0, false, false);
        acc1 = __builtin_amdgcn_wmma_f32_16x16x4_f32(false, a, false, b1,
                                                     (short)0, acc1, false, false);
        acc2 = __builtin_amdgcn_wmma_f32_16x16x4_f32(false, a, false, b2,
                                                     (short)0, acc2, false, false);
        acc3 = __builtin_amdgcn_wmma_f32_16x16x4_f32(false, a, false, b3,
                                                     (short)0, acc3, false, false);
      }
    }
  }
  if (m0 < nrows) {
#pragma unroll
    for (int r = 0; r < 8; ++r) {
      const int row = m0 + r + 8 * khalf;
      float* crow = &C[(size_t)row * 64 + mrow];
      crow[0]  = acc0[r];
      crow[16] = acc1[r];
      crow[32] = acc2[r];
      crow[48] = acc3[r];
    }
  }
}

// ------------------------- attention dot products --------------------------

// layer 1: a_s[n,h] = sum_f h[n,h,f]*att_src[h,f]   (8 heads x 8 dims)
__global__ __launch_bounds__(256) void attn_scores_h8(
    const float* __restrict__ h, const float* __restrict__ asrc,
    const float* __restrict__ adst, float* __restrict__ os,
    float* __restrict__ od) {
  int tid = blockIdx.x * 256 + threadIdx.x;
  if (tid >= Nn * H1) return;
  int n = tid >> 3, hd = tid & 7;
  const float* hp = &h[(size_t)n * HID + hd * F1];
  float s = 0.0f, d = 0.0f;
#pragma unroll
  for (int f = 0; f < F1; ++f) {
    s += hp[f] * asrc[hd * F1 + f];
    d += hp[f] * adst[hd * F1 + f];
  }
  os[tid] = s;
  od[tid] = d;
}

// layer 2: single head over 64 dims
__global__ __launch_bounds__(256) void attn_scores_h1(
    const float* __restrict__ h, const float* __restrict__ asrc,
    const float* __restrict__ adst, float* __restrict__ os,
    float* __restrict__ od) {
  int n = blockIdx.x * 256 + threadIdx.x;
  if (n >= Nn) return;
  const float* hp = &h[(size_t)n * OUT];
  float s = 0.0f, d = 0.0f;
  for (int f = 0; f < OUT; ++f) {
    s += hp[f] * asrc[f];
    d += hp[f] * adst[f];
  }
  os[n] = s;
  od[n] = d;
}

// --------------------------- edge softmax passes ---------------------------

template <int H>
__global__ __launch_bounds__(256) void edge_softmax_max(
    const int* __restrict__ ei, const float* __restrict__ as,
    const float* __restrict__ ad, unsigned* __restrict__ mx) {
  int tid = blockIdx.x * 256 + threadIdx.x;
  if (tid >= Eall * H) return;
  int e = tid / H, h = tid % H;
  int s, d;
  edge_ends(ei, e, s, d);
  float a = leaky(as[s * H + h] + ad[d * H + h]);
  atomicMax(&mx[d * H + h], fenc(a));
}

template <int H>
__global__ __launch_bounds__(256) void edge_softmax_denom(
    const int* __restrict__ ei, const float* __restrict__ as,
    const float* __restrict__ ad, const unsigned* __restrict__ mx,
    float* __restrict__ denom) {
  int tid = blockIdx.x * 256 + threadIdx.x;
  if (tid >= Eall * H) return;
  int e = tid / H, h = tid % H;
  int s, d;
  edge_ends(ei, e, s, d);
  float a  = leaky(as[s * H + h] + ad[d * H + h]);
  float ex = expf(a - fdec(mx[d * H + h]));
  atomicAdd(&denom[d * H + h], ex);
}

// one thread per (edge, feature); feature in the low bits -> coalesced
// gather of h[src] and coalesced atomics on agg[dst].
template <int H, int F>
__global__ __launch_bounds__(256) void edge_aggregate(
    const int* __restrict__ ei, const float* __restrict__ as,
    const float* __restrict__ ad, const unsigned* __restrict__ mx,
    const float* __restrict__ denom, const float* __restrict__ feat,
    float* __restrict__ agg) {
  int tid = blockIdx.x * 256 + threadIdx.x;
  if (tid >= Eall * 64) return;
  int e = tid >> 6, c = tid & 63;
  int h = c / F;                      // F is a template constant -> shift
  int s, d;
  edge_ends(ei, e, s, d);
  float a    = leaky(as[s * H + h] + ad[d * H + h]);
  float ex   = expf(a - fdec(mx[d * H + h]));
  float coef = ex / (denom[d * H + h] + 1e-16f);
  atomicAdd(&agg[(size_t)d * 64 + c], feat[(size_t)s * 64 + c] * coef);
}

// --------------------------- pointwise epilogues ---------------------------

__global__ __launch_bounds__(256) void bias_elu(float* __restrict__ a,
                                                const float* __restrict__ b) {
  int i = blockIdx.x * 256 + threadIdx.x;
  if (i >= Nn * HID) return;
  float v = a[i] + b[i & 63];
  a[i] = v > 0.0f ? v : expm1f(v);
}

__global__ __launch_bounds__(256) void bias_logsoftmax(
    float* __restrict__ out, const float* __restrict__ b) {
  int n = blockIdx.x * 256 + threadIdx.x;
  if (n >= Nn) return;
  float v[OUT];
  float m = -INFINITY;
#pragma unroll
  for (int c = 0; c < OUT; ++c) {
    v[c] = out[(size_t)n * OUT + c] + b[c];
    m = fmaxf(m, v[c]);
  }
  float sum = 0.0f;
#pragma unroll
  for (int c = 0; c < OUT; ++c) sum += expf(v[c] - m);
  float lse = m + logf(sum);
#pragma unroll
  for (int c = 0; c < OUT; ++c) out[(size_t)n * OUT + c] = v[c] - lse;
}

// ------------------------------- launcher ----------------------------------

extern "C" void kernel_launch(void* const* d_in, const int* in_sizes, int n_in,
                              void* d_out, int out_size, void* d_ws,
                              size_t ws_size, hipStream_t stream) {
  (void)in_sizes; (void)n_in; (void)out_size; (void)ws_size;

  const float* x        = (const float*)d_in[0];
  const int*   ei       = (const int*)  d_in[1];   // [2, E]
  const float* W1       = (const float*)d_in[2];
  const float* att_src1 = (const float*)d_in[3];
  const float* att_dst1 = (const float*)d_in[4];
  const float* b1       = (const float*)d_in[5];
  const float* W2       = (const float*)d_in[6];
  const float* att_src2 = (const float*)d_in[7];
  const float* att_dst2 = (const float*)d_in[8];
  const float* b2       = (const float*)d_in[9];
  float* out = (float*)d_out;

  // workspace layout (floats); 160*N*4 = 32 MB total
  float* f = (float*)d_ws;
  float*    h1   = f;                         // N*64  layer-1 features (reused as layer-2 features)
  float*    agg1 = f + (size_t)Nn * 64;       // N*64  layer-1 aggregate -> ELU -> GEMM2 input
  float*    as1  = f + (size_t)Nn * 128;      // N*8   (layer 2 reuses first N)
  float*    ad1  = f + (size_t)Nn * 136;      // N*8
  unsigned* mx1  = (unsigned*)(f + (size_t)Nn * 144);  // N*8
  float*    dn1  = f + (size_t)Nn * 152;      // N*8

  const int tiles  = Nn / 16;                       // 3125
  const int gblk   = (tiles + 7) / 8;               // 8 waves / block
  auto blks = [](long long n) { return (int)((n + 255) / 256); };

  // ---- layer 1 ----
  zero_f32<<<blks((long long)Nn * 96), 256, 0, stream>>>(agg1, Nn * 96); // agg1+as1+ad1+mx1+dn1
  gemm_wmma_f32<IN><<<gblk, 256, 0, stream>>>(x, W1, h1, Nn);
  attn_scores_h8<<<blks((long long)Nn * H1), 256, 0, stream>>>(h1, att_src1, att_dst1, as1, ad1);
  edge_softmax_max<H1><<<blks((long long)Eall * H1), 256, 0, stream>>>(ei, as1, ad1, mx1);
  edge_softmax_denom<H1><<<blks((long long)Eall * H1), 256, 0, stream>>>(ei, as1, ad1, mx1, dn1);
  edge_aggregate<H1, F1><<<blks((long long)Eall * 64), 256, 0, stream>>>(
      ei, as1, ad1, mx1, dn1, h1, agg1);
  bias_elu<<<blks((long long)Nn * HID), 256, 0, stream>>>(agg1, b1);

  // ---- layer 2 (reuse buffers; re-zero the state it needs) ----
  zero_f32<<<blks((long long)Nn), 256, 0, stream>>>((float*)mx1, Nn);
  zero_f32<<<blks((long long)Nn), 256, 0, stream>>>(dn1, Nn);
  zero_f32<<<blks((long long)Nn * OUT), 256, 0, stream>>>(out, Nn * OUT);
  gemm_wmma_f32<HID><<<gblk, 256, 0, stream>>>(agg1, W2, h1, Nn);
  attn_scores_h1<<<blks((long long)Nn), 256, 0, stream>>>(h1, att_src2, att_dst2, as1, ad1);
  edge_softmax_max<1><<<blks((long long)Eall), 256, 0, stream>>>(ei, as1, ad1, mx1);
  edge_softmax_denom<1><<<blks((long long)Eall), 256, 0, stream>>>(ei, as1, ad1, mx1, dn1);
  edge_aggregate<1, OUT><<<blks((long long)Eall * 64), 256, 0, stream>>>(
      ei, as1, ad1, mx1, dn1, h1, out);
  bias_logsoftmax<<<blks((long long)Nn), 256, 0, stream>>>(out, b2);
}